// HeteroGNN_16904991277356
// MI455X (gfx1250) — compile-verified
//
#include <hip/hip_runtime.h>

typedef __attribute__((ext_vector_type(2))) float v2f;
typedef __attribute__((ext_vector_type(8))) float v8f;

#define CH    128
#define RNUM  4
#define NMAT  5   // 4 relations + root transform
#define TCONV 4

// One block = 8 waves; wave w computes the 16-row strip (blockIdx*8+w) of
// h[n, m, :] for all 5 matrices.  Each 128x128 weight matrix is staged into
// LDS once per block (64 KB) in K-PAIR layout: ldsWp[(k/2)*128 + col] holds
// {W[k][col], W[k+1][col]}, so each WMMA B operand is one aligned b64 LDS
// load.  Two col-tiles per pass give two independent accumulator chains.
__global__ __launch_bounds__(256) void rgcn_gemm(
    const float* __restrict__ x,     // [N, 128]
    const float* __restrict__ W,     // [4, 128, 128] (this conv's relations)
    const float* __restrict__ root,  // [128, 128]
    float* __restrict__ h,           // [N, 5, 128]
    int N, int nstrips)
{
    __shared__ float2 ldsWp[64 * CH];  // 64 KB (WGP has 320 KB)

    const int wave = blockIdx.x * (blockDim.x >> 5) + (threadIdx.x >> 5);
    const bool wave_active = (wave < nstrips);   // wave-uniform
    const int lane = threadIdx.x & 31;
    const int half = lane >> 4;                  // lane half selects K pair
    const int mrow = lane & 15;
    const int row_base = wave * 16;

    // A 16x4 f32 layout: lanes 0-15 rows M=0..15 with VGPR0=K0,VGPR1=K1;
    // lanes 16-31 same rows with K2,K3.  Keep whole 16x128 strip in regs.
    v2f a[32];
    {
        int arow_idx = row_base + mrow;
        if (arow_idx >= N) arow_idx = N - 1;     // clamp (harmless over-read)
        if (!wave_active) arow_idx = 0;
        const float* arow = x + (size_t)arow_idx * CH;
#pragma unroll
        for (int kk = 0; kk < 32; ++kk) {
            const int col = kk * 4 + half * 2;
            a[kk].x = arow[col];
            a[kk].y = arow[col + 1];
        }
    }

    for (int m = 0; m < NMAT; ++m) {
        const float* Wm = (m < RNUM) ? (W + (size_t)m * CH * CH) : root;

        // cooperative stage into K-pair layout: 2048 2x4 chunks, 8 per thread.
        // chunk (kp, cg): rows 2kp,2kp+1 x cols 4cg..4cg+3 -> two b128 stores.
        __syncthreads();
        {
            const float4* Wm4 = (const float4*)Wm;
#pragma unroll
            for (int i = 0; i < 8; ++i) {
                const int idx = i * 256 + threadIdx.x;  // 0..2047
                const int kp = idx >> 5;                // K pair 0..63
                const int cg = idx & 31;                // col group of 4
                const float4 r0 = Wm4[(kp * 2) * 32 + cg];
                const float4 r1 = Wm4[(kp * 2 + 1) * 32 + cg];
                float4* dst = (float4*)&ldsWp[kp * CH + cg * 4];
                dst[0] = make_float4(r0.x, r1.x, r0.y, r1.y);
                dst[1] = make_float4(r0.z, r1.z, r0.w, r1.w);
            }
        }
        __syncthreads();

        if (wave_active) {
            for (int cp = 0; cp < 4; ++cp) {     // two col-tiles per pass
                const int col0 = cp * 32 + mrow;
                v8f acc0 = {};
                v8f acc1 = {};
#pragma unroll
                for (int kk = 0; kk < 32; ++kk) {
                    // K pair index for this lane-half = k0/2 = kk*2 + half
                    const float2* bp = &ldsWp[(kk * 2 + half) * CH + col0];
                    const float2 t0 = bp[0];
                    const float2 t1 = bp[16];
                    v2f b0; b0.x = t0.x; b0.y = t0.y;
                    v2f b1; b1.x = t1.x; b1.y = t1.y;
                    acc0 = __builtin_amdgcn_wmma_f32_16x16x4_f32(
                        false, a[kk], false, b0, (short)0, acc0, false, false);
                    acc1 = __builtin_amdgcn_wmma_f32_16x16x4_f32(
                        false, a[kk], false, b1, (short)0, acc1, false, false);
                }
                // C/D layout: VGPR v -> row (v + 8*half), lane -> col
#pragma unroll
                for (int v = 0; v < 8; ++v) {
                    const int node = row_base + v + 8 * half;
                    if (node < N) {
                        float* hn = h + ((size_t)node * NMAT + m) * CH + mrow;
                        hn[cp * 32]      = acc0[v];
                        hn[cp * 32 + 16] = acc1[v];
                    }
                }
            }
        }
    }
}

__global__ void zero_i32(int* __restrict__ p, int n)
{
    int i = blockIdx.x * blockDim.x + threadIdx.x;
    if (i < n) p[i] = 0;
}

__global__ void count_edges(const int* __restrict__ dst, const int* __restrict__ et,
                            int* __restrict__ cnt, int E)
{
    int e = blockIdx.x * blockDim.x + threadIdx.x;
    if (e < E) atomicAdd(&cnt[dst[e] * RNUM + et[e]], 1);
}

__global__ void init_out(const float* __restrict__ h, const float* __restrict__ bias,
                         float* __restrict__ out, int total)
{
    int i = blockIdx.x * blockDim.x + threadIdx.x;
    if (i >= total) return;
    const int n = i >> 7, c = i & 127;
    out[i] = h[((size_t)n * NMAT + RNUM) * CH + c] + bias[c];
}

// One 32-lane group per edge; each lane moves 4 channels (float4) -> 4 f32 atomics.
__global__ void scatter_edges(const float* __restrict__ h,
                              const int* __restrict__ src, const int* __restrict__ dst,
                              const int* __restrict__ et, const int* __restrict__ cnt,
                              float* __restrict__ out, int E)
{
    int tid = blockIdx.x * blockDim.x + threadIdx.x;
    int e = tid >> 5;
    if (e >= E) return;
    const int lane = tid & 31;
    const int s = src[e], d = dst[e], r = et[e];
    const int c = cnt[d * RNUM + r];
    const float w = 1.0f / (float)(c > 1 ? c : 1);
    const float* hp = h + ((size_t)s * NMAT + r) * CH + lane * 4;
    const float4 v = *(const float4*)hp;
    float* o = out + (size_t)d * CH + lane * 4;
    atomicAdd(o + 0, v.x * w);
    atomicAdd(o + 1, v.y * w);
    atomicAdd(o + 2, v.z * w);
    atomicAdd(o + 3, v.w * w);
}

extern "C" void kernel_launch(void* const* d_in, const int* in_sizes, int n_in,
                              void* d_out, int out_size, void* d_ws, size_t ws_size,
                              hipStream_t stream)
{
    const float* x_in       = (const float*)d_in[0];  // [N,128]
    const float* W_all      = (const float*)d_in[1];  // [T,4,128,128]
    const float* root_all   = (const float*)d_in[2];  // [T,128,128]
    const float* bias_all   = (const float*)d_in[3];  // [T,128]
    const int*   edge_index = (const int*)d_in[4];    // [T,2,E]
    const int*   edge_type  = (const int*)d_in[5];    // [T,E]

    const int N = in_sizes[0] / CH;
    const int E = in_sizes[5] / TCONV;

    // workspace layout
    float* h  = (float*)d_ws;                      // N*5*128 floats (256 MB)
    float* x0 = h + (size_t)N * NMAT * CH;         // N*128
    float* x1 = x0 + (size_t)N * CH;               // N*128
    int*   cnt = (int*)(x1 + (size_t)N * CH);      // N*4
    float* out = (float*)d_out;

    const int nstrips = (N + 15) / 16;
    const int gemm_blocks = (nstrips + 7) / 8;     // 8 waves / 256-thread block
    const int cnt_n = N * RNUM;
    const int out_n = N * CH;
    const long scat_threads = (long)E * 32;

    const float* cur = x_in;
    for (int t = 0; t < TCONV; ++t) {
        float* nxt = (t == TCONV - 1) ? out : ((t & 1) ? x1 : x0);
        const float* Wt    = W_all    + (size_t)t * RNUM * CH * CH;
        const float* roott = root_all + (size_t)t * CH * CH;
        const float* biast = bias_all + (size_t)t * CH;
        const int* srcp = edge_index + (size_t)t * 2 * E;
        const int* dstp = srcp + E;
        const int* etp  = edge_type + (size_t)t * E;

        rgcn_gemm<<<gemm_blocks, 256, 0, stream>>>(cur, Wt, roott, h, N, nstrips);
        zero_i32<<<(cnt_n + 255) / 256, 256, 0, stream>>>(cnt, cnt_n);
        count_edges<<<(E + 255) / 256, 256, 0, stream>>>(dstp, etp, cnt, E);
        init_out<<<(out_n + 255) / 256, 256, 0, stream>>>(h, biast, nxt, out_n);
        scatter_edges<<<(int)((scat_threads + 255) / 256), 256, 0, stream>>>(
            h, srcp, dstp, etp, cnt, nxt, E);
        cur = nxt;
    }
}